// MultiHeadAttention_TUPE_74165495267705
// MI455X (gfx1250) — compile-verified
//
#include <hip/hip_runtime.h>
#include <math.h>

// ---------------------------------------------------------------------------
// CDNA5 (gfx1250) TUPE attention, fp32 WMMA pipeline:
//   qpe = pe @ Uq ; kpe = pe @ Uk                 (GEMM, B is KxN)
//   attn_raw[b] = q[b] @ k[b]^T                   (GEMM, B is NxK, batched z=4)
//   attn_raw[b] = (attn_raw[b] + qpe@kpe^T)/sqrt(128)  (GEMM w/ broadcast epilogue)
//   softmax rows in place (attn is a required output -> no flash attention)
//   ctx[b] = (attn[b] @ v[b]) stored d-major      (transposed-store epilogue)
//
// GEMM: 128x64 block tile, BK=16, 8 waves (4x2), each wave owns a 32x32 tile
// (4 x v8f accumulators, outer-product A/B fragment reuse). Fragments are
// 8B-aligned v2f LDS loads matching the 32-bit WMMA operand layout
// (lane<16 -> K={0,1}, lane>=16 -> K={2,3}).
// ---------------------------------------------------------------------------

typedef __attribute__((ext_vector_type(2))) float v2f;
typedef __attribute__((ext_vector_type(8))) float v8f;

#define TM 128
#define TN 64
#define BK 16
#define LDS_STRIDE 20  // keeps float4 stores 16B aligned, breaks bank conflicts

__device__ __forceinline__ v8f wmma_f32_k4(v2f a, v2f b, v8f c) {
  return __builtin_amdgcn_wmma_f32_16x16x4_f32(
      /*neg_a=*/false, a, /*neg_b=*/false, b,
      /*c_mod=*/(short)0, c, /*reuse_a=*/false, /*reuse_b=*/false);
}

// EPI: 0 = C[m][n] = acc
//      2 = C[n][m] = acc                      (transposed store)
//      3 = for bb in 0..3: C[bb][m][n] = (C[bb][m][n] + acc) * scale
// BNK: true  -> B given as N x K rows (B^T product; load rows directly)
//      false -> B given as K x N (transpose while staging into LDS)
template <bool BNK, int EPI>
__global__ __launch_bounds__(256) void gemm_wmma(
    const float* __restrict__ A, long long aBatch, int lda,
    const float* __restrict__ B, long long bBatch, int ldb,
    float* __restrict__ C, long long cBatch, int ldc,
    int K, float scale) {
  __shared__ float As[TM * LDS_STRIDE];
  __shared__ float Bs[TN * LDS_STRIDE];

  const int z = blockIdx.z;
  A += (long long)z * aBatch;
  B += (long long)z * bBatch;
  if (EPI != 3) C += (long long)z * cBatch;

  const int bm = blockIdx.y * TM;
  const int bn = blockIdx.x * TN;

  const int t    = threadIdx.x;
  const int w    = t >> 5;      // wave 0..7 (wave32)
  const int lane = t & 31;
  const int half = lane >> 4;   // K-half select for 32-bit fragments
  const int nm   = lane & 15;   // row (A) / col (B) index within 16

  const int tm0 = (w & 3) * 32;   // wave's 32-row slab
  const int tn0 = (w >> 2) * 32;  // wave's 32-col slab

  v8f acc00 = {}, acc01 = {}, acc10 = {}, acc11 = {};

  // staging: A tile 128x16 (two 64-row passes), B tile 64x16
  const int la_r = t >> 2;        // 0..63
  const int la_c = (t & 3) * 4;   // 0,4,8,12
  const int lb_k = t >> 4;        // 0..15 (BNK=false)
  const int lb_n = (t & 15) * 4;  // 0..60 (BNK=false)

  for (int k0 = 0; k0 < K; k0 += BK) {
    const float4 av0 = *(const float4*)&A[(long long)(bm + la_r) * lda + (k0 + la_c)];
    const float4 av1 = *(const float4*)&A[(long long)(bm + la_r + 64) * lda + (k0 + la_c)];
    float4 bv;
    if (BNK) {
      bv = *(const float4*)&B[(long long)(bn + la_r) * ldb + (k0 + la_c)];
    } else {
      bv = *(const float4*)&B[(long long)(k0 + lb_k) * ldb + (bn + lb_n)];
    }
    __syncthreads();  // protect previous iteration's LDS reads
    *(float4*)&As[la_r * LDS_STRIDE + la_c] = av0;
    *(float4*)&As[(la_r + 64) * LDS_STRIDE + la_c] = av1;
    if (BNK) {
      *(float4*)&Bs[la_r * LDS_STRIDE + la_c] = bv;
    } else {
      Bs[(lb_n + 0) * LDS_STRIDE + lb_k] = bv.x;
      Bs[(lb_n + 1) * LDS_STRIDE + lb_k] = bv.y;
      Bs[(lb_n + 2) * LDS_STRIDE + lb_k] = bv.z;
      Bs[(lb_n + 3) * LDS_STRIDE + lb_k] = bv.w;
    }
    __syncthreads();

#pragma unroll
    for (int kk = 0; kk < 4; ++kk) {
      const int kb = kk * 4 + half * 2;  // even -> 8B-aligned ds_load_b64
      const v2f a0 = *(const v2f*)&As[(tm0 + nm) * LDS_STRIDE + kb];
      const v2f a1 = *(const v2f*)&As[(tm0 + 16 + nm) * LDS_STRIDE + kb];
      const v2f b0 = *(const v2f*)&Bs[(tn0 + nm) * LDS_STRIDE + kb];
      const v2f b1 = *(const v2f*)&Bs[(tn0 + 16 + nm) * LDS_STRIDE + kb];
      acc00 = wmma_f32_k4(a0, b0, acc00);
      acc01 = wmma_f32_k4(a0, b1, acc01);
      acc10 = wmma_f32_k4(a1, b0, acc10);
      acc11 = wmma_f32_k4(a1, b1, acc11);
    }
  }

  // 16x16 f32 C/D layout: element (m, n) -> vgpr = m & 7, lane = n + 16*(m >> 3)
#pragma unroll
  for (int e = 0; e < 8; ++e) {
    const int mg0 = bm + tm0 + half * 8 + e;
    const int mg1 = mg0 + 16;
    const int ng0 = bn + tn0 + nm;
    const int ng1 = ng0 + 16;
    const float c00 = acc00[e], c01 = acc01[e];
    const float c10 = acc10[e], c11 = acc11[e];
    if (EPI == 2) {
      C[(long long)ng0 * ldc + mg0] = c00;
      C[(long long)ng1 * ldc + mg0] = c01;
      C[(long long)ng0 * ldc + mg1] = c10;
      C[(long long)ng1 * ldc + mg1] = c11;
    } else if (EPI == 3) {
      const long long i00 = (long long)mg0 * ldc + ng0;
      const long long i01 = (long long)mg0 * ldc + ng1;
      const long long i10 = (long long)mg1 * ldc + ng0;
      const long long i11 = (long long)mg1 * ldc + ng1;
#pragma unroll
      for (int bb = 0; bb < 4; ++bb) {
        const long long off = (long long)bb * cBatch;
        C[i00 + off] = (C[i00 + off] + c00) * scale;
        C[i01 + off] = (C[i01 + off] + c01) * scale;
        C[i10 + off] = (C[i10 + off] + c10) * scale;
        C[i11 + off] = (C[i11 + off] + c11) * scale;
      }
    } else {
      C[(long long)mg0 * ldc + ng0] = c00;
      C[(long long)mg0 * ldc + ng1] = c01;
      C[(long long)mg1 * ldc + ng0] = c10;
      C[(long long)mg1 * ldc + ng1] = c11;
    }
  }
}

// Row softmax, one 256-thread block per row of 2048, in place.
__global__ __launch_bounds__(256) void softmax_rows(float* __restrict__ p2, int cols) {
  __shared__ float red[8];
  float* p = p2 + (long long)blockIdx.x * cols;
  const int t = threadIdx.x;

  float vals[8];
  float m = -3.402823466e38f;
#pragma unroll
  for (int i = 0; i < 8; ++i) {
    vals[i] = p[t + i * 256];
    m = fmaxf(m, vals[i]);
  }
#pragma unroll
  for (int off = 16; off >= 1; off >>= 1) m = fmaxf(m, __shfl_xor(m, off, 32));
  if ((t & 31) == 0) red[t >> 5] = m;
  __syncthreads();
  float bm = red[0];
#pragma unroll
  for (int i = 1; i < 8; ++i) bm = fmaxf(bm, red[i]);

  float s = 0.0f;
#pragma unroll
  for (int i = 0; i < 8; ++i) {
    vals[i] = expf(vals[i] - bm);
    s += vals[i];
  }
#pragma unroll
  for (int off = 16; off >= 1; off >>= 1) s += __shfl_xor(s, off, 32);
  __syncthreads();  // done reading red before rewrite
  if ((t & 31) == 0) red[t >> 5] = s;
  __syncthreads();
  float bs = 0.0f;
#pragma unroll
  for (int i = 0; i < 8; ++i) bs += red[i];
  const float inv = 1.0f / bs;
#pragma unroll
  for (int i = 0; i < 8; ++i) p[t + i * 256] = vals[i] * inv;
}

extern "C" void kernel_launch(void* const* d_in, const int* in_sizes, int n_in,
                              void* d_out, int out_size, void* d_ws, size_t ws_size,
                              hipStream_t stream) {
  (void)in_sizes; (void)n_in; (void)out_size; (void)d_ws; (void)ws_size;

  const float* q  = (const float*)d_in[0];  // [4,2048,1024]
  const float* k  = (const float*)d_in[1];  // [4,2048,1024]
  const float* v  = (const float*)d_in[2];  // [4,2048,1024]
  const float* pe = (const float*)d_in[3];  // [10000,1024] (first 2048 rows used)
  const float* Uq = (const float*)d_in[4];  // [1024,1024]  (K x N)
  const float* Uk = (const float*)d_in[5];  // [1024,1024]

  const int B = 4, S = 2048, D = 1024;
  float* out_ctx  = (float*)d_out;                         // B*S*D (d-major per batch)
  float* out_attn = (float*)d_out + (long long)B * S * D;  // B*S*S

  // qpe/kpe scratch lives in the ctx region: dead before the context GEMM
  // overwrites it (2*S*D <= B*S*D). Zero dependence on ws_size.
  float* qpe = out_ctx;
  float* kpe = out_ctx + (long long)S * D;

  const float inv_scale = 0.08838834764831845f;  // 1/sqrt(2*64)
  const dim3 blk(256);

  // 1) pe projections: [2048x1024] = pe[2048x1024] @ U[1024x1024]  (B is KxN)
  const dim3 g1(D / TN, S / TM, 1);
  gemm_wmma<false, 0><<<g1, blk, 0, stream>>>(pe, 0, D, Uq, 0, D, qpe, 0, D, D, 1.0f);
  gemm_wmma<false, 0><<<g1, blk, 0, stream>>>(pe, 0, D, Uk, 0, D, kpe, 0, D, D, 1.0f);

  // 2) raw scores, batched: attn_raw[b] = q[b] @ k[b]^T  (B is NxK)
  const dim3 g2(S / TN, S / TM, B);
  gemm_wmma<true, 0><<<g2, blk, 0, stream>>>(q, (long long)S * D, D,
                                             k, (long long)S * D, D,
                                             out_attn, (long long)S * S, S, D, 1.0f);

  // 3) bias GEMM w/ broadcast epilogue: attn_raw[b] = (attn_raw[b] + qpe@kpe^T) * inv_scale
  const dim3 g3(S / TN, S / TM, 1);
  gemm_wmma<true, 3><<<g3, blk, 0, stream>>>(qpe, 0, D, kpe, 0, D,
                                             out_attn, (long long)S * S, S, D, inv_scale);

  // 4) softmax rows in place (B*S rows of S)
  softmax_rows<<<dim3(B * S), blk, 0, stream>>>(out_attn, S);

  // 5) context, batched, transposed store: ctx[b][d][q] = (attn[b] @ v[b])[q][d]
  const dim3 g5(D / TN, S / TM, B);
  gemm_wmma<false, 2><<<g5, blk, 0, stream>>>(out_attn, (long long)S * S, S,
                                              v, (long long)S * D, D,
                                              out_ctx, (long long)S * D, S, S, 1.0f);
}